// GATLayerV2_51994874085817
// MI455X (gfx1250) — compile-verified
//
#include <hip/hip_runtime.h>
#include <hip/hip_bf16.h>
#include <math.h>

// GAT layer for MI455X (gfx1250). fp32 WMMA for the x@W GEMM, L2-resident
// gather/scatter for the edge stages (working set ~40MB << 192MB L2).

typedef float v2f __attribute__((ext_vector_type(2)));
typedef float v8f __attribute__((ext_vector_type(8)));

#define NEG_SLOPE 0.2f

__device__ __forceinline__ float lrelu(float v) {
    return v > 0.f ? v : NEG_SLOPE * v;
}

// Exact float atomic-max with init = -inf:
//   nonneg values: signed-int max (positive float bits order like ints)
//   negative values: unsigned-int min (more-negative floats have larger bits)
__device__ __forceinline__ void atomicMaxF(float* addr, float v) {
    if (v >= 0.f) atomicMax((int*)addr, __float_as_int(v));
    else          atomicMin((unsigned int*)addr, __float_as_uint(v));
}

// ---------------------------------------------------------------------------
// K1: Wx = x @ W  via V_WMMA_F32_16X16X4_F32.
// One wave per 16-row tile. A is 16x4 fp32 per step (lanes 0-15: K=0,1;
// lanes 16-31: K=2,3 per the ISA 32-bit A layout). Two accumulators cover
// the 32 output columns. K=64 -> 16 WMMA steps x 2 tiles = 32 wmma/wave.
// ---------------------------------------------------------------------------
__global__ __launch_bounds__(32) void gat_gemm_wmma(
    const float* __restrict__ x, const float* __restrict__ W,
    float* __restrict__ Wx, int N)
{
    const int lane = threadIdx.x & 31;
    const int half = lane >> 4;      // 0: K even pair, 1: K odd pair
    const int l    = lane & 15;
    const int rowBase = blockIdx.x << 4;
    int row = rowBase + l;
    if (row >= N) row = N - 1;       // clamp loads; EXEC stays all-ones

    v8f acc0 = {0.f,0.f,0.f,0.f,0.f,0.f,0.f,0.f};
    v8f acc1 = {0.f,0.f,0.f,0.f,0.f,0.f,0.f,0.f};

    const float* xr = x + (size_t)row * 64 + half * 2;
#pragma unroll
    for (int c = 0; c < 16; ++c) {
        v2f a = *(const v2f*)(xr + 4 * c);          // K = 4c + 2*half .. +1
        const int k0 = 4 * c + 2 * half;
        v2f b0, b1;                                  // B: 4x16, row striped across lanes
        b0.x = W[(k0 + 0) * 32 + l];
        b0.y = W[(k0 + 1) * 32 + l];
        b1.x = W[(k0 + 0) * 32 + 16 + l];
        b1.y = W[(k0 + 1) * 32 + 16 + l];
        acc0 = __builtin_amdgcn_wmma_f32_16x16x4_f32(false, a, false, b0,
                                                     (short)0, acc0, false, false);
        acc1 = __builtin_amdgcn_wmma_f32_16x16x4_f32(false, a, false, b1,
                                                     (short)0, acc1, false, false);
    }

    // C/D layout: VGPR r holds M=r (lanes 0-15) / M=r+8 (lanes 16-31), N=l.
#pragma unroll
    for (int r = 0; r < 8; ++r) {
        const int m = rowBase + r + 8 * half;
        if (m < N) {
            Wx[(size_t)m * 32 + l]      = acc0[r];
            Wx[(size_t)m * 32 + 16 + l] = acc1[r];
        }
    }
}

// ---------------------------------------------------------------------------
// K2: per-node attention scalars + init of segment-softmax state + zero h.
// ---------------------------------------------------------------------------
__global__ void gat_node_prep(
    const float* __restrict__ Wx,
    const float* __restrict__ att_src, const float* __restrict__ att_dst,
    float* __restrict__ s_src, float* __restrict__ s_dst,
    float* __restrict__ e_max, float* __restrict__ sum_exp,
    float* __restrict__ h, int N)
{
    int n = blockIdx.x * blockDim.x + threadIdx.x;
    if (n >= N) return;
    const float* w = Wx + (size_t)n * 32;
    float ss = 0.f, sd = 0.f;
#pragma unroll
    for (int k = 0; k < 32; ++k) {
        float v = w[k];
        ss = fmaf(v, att_src[k], ss);
        sd = fmaf(v, att_dst[k], sd);
    }
    s_src[n]   = ss;
    s_dst[n]   = sd;
    e_max[n]   = -INFINITY;
    sum_exp[n] = 0.f;
    float4 z = make_float4(0.f, 0.f, 0.f, 0.f);
    float4* hp = (float4*)(h + (size_t)n * 32);
#pragma unroll
    for (int k = 0; k < 8; ++k) hp[k] = z;
}

// ---------------------------------------------------------------------------
// K3: per-edge score + segment max over dst.
// ---------------------------------------------------------------------------
__global__ void gat_edge_max(
    const int* __restrict__ src, const int* __restrict__ dst,
    const float* __restrict__ s_src, const float* __restrict__ s_dst,
    float* __restrict__ e_max, int E)
{
    int e = blockIdx.x * blockDim.x + threadIdx.x;
    if (e >= E) return;
    int s = src[e], d = dst[e];
    float v = lrelu(s_dst[d] + s_src[s]);
    atomicMaxF(&e_max[d], v);
}

// ---------------------------------------------------------------------------
// K4: per-edge exp(e - max) -> segment sum (GLOBAL_ATOMIC_ADD_F32).
// ---------------------------------------------------------------------------
__global__ void gat_edge_sum(
    const int* __restrict__ src, const int* __restrict__ dst,
    const float* __restrict__ s_src, const float* __restrict__ s_dst,
    const float* __restrict__ e_max, float* __restrict__ sum_exp, int E)
{
    int e = blockIdx.x * blockDim.x + threadIdx.x;
    if (e >= E) return;
    int s = src[e], d = dst[e];
    float v  = lrelu(s_dst[d] + s_src[s]);
    float ex = __expf(v - e_max[d]);
    atomicAdd(&sum_exp[d], ex);
}

// ---------------------------------------------------------------------------
// K5: one wave per edge; lane k handles output column k.
// Scatter alpha * Wx[src] into h[dst] with f32 atomics (consecutive lanes
// hit consecutive addresses of one dst row -> coalesced L2 atomics).
// ---------------------------------------------------------------------------
__global__ void gat_edge_scatter(
    const int* __restrict__ src, const int* __restrict__ dst,
    const float* __restrict__ s_src, const float* __restrict__ s_dst,
    const float* __restrict__ e_max, const float* __restrict__ sum_exp,
    const float* __restrict__ Wx, float* __restrict__ h, int E)
{
    unsigned long long t =
        (unsigned long long)blockIdx.x * blockDim.x + threadIdx.x;
    int e = (int)(t >> 5);
    int k = (int)(t & 31);
    if (e >= E) return;
    int s = src[e], d = dst[e];   // uniform within the wave -> scalarized
    float v     = lrelu(s_dst[d] + s_src[s]);
    float ex    = __expf(v - e_max[d]);
    float alpha = ex / (sum_exp[d] + 1e-16f);
    atomicAdd(&h[(size_t)d * 32 + k], alpha * Wx[(size_t)s * 32 + k]);
}

// ---------------------------------------------------------------------------
// K6: in-place ELU on the accumulated output.
// ---------------------------------------------------------------------------
__global__ void gat_elu(float* __restrict__ h, int total)
{
    int i = blockIdx.x * blockDim.x + threadIdx.x;
    if (i >= total) return;
    float v = h[i];
    h[i] = v > 0.f ? v : (__expf(v) - 1.f);
}

// ---------------------------------------------------------------------------
extern "C" void kernel_launch(void* const* d_in, const int* in_sizes, int n_in,
                              void* d_out, int out_size, void* d_ws, size_t ws_size,
                              hipStream_t stream)
{
    const float* x       = (const float*)d_in[0];   // [N, 64]
    const int*   ei      = (const int*)d_in[1];     // [2, E]
    const float* W       = (const float*)d_in[2];   // [64, 32]
    const float* att_src = (const float*)d_in[3];   // [32]
    const float* att_dst = (const float*)d_in[4];   // [32]
    float*       out     = (float*)d_out;           // [N, 32]

    const int N = in_sizes[0] / 64;
    const int E = in_sizes[1] / 2;
    const int* src = ei;
    const int* dst = ei + E;

    // Workspace layout (floats): Wx[N*32] | s_src[N] | s_dst[N] | e_max[N] | sum_exp[N]
    float* ws      = (float*)d_ws;
    float* Wx      = ws;
    float* s_src   = Wx + (size_t)N * 32;
    float* s_dst   = s_src + N;
    float* e_max   = s_dst + N;
    float* sum_exp = e_max + N;

    const int row_tiles = (N + 15) / 16;
    gat_gemm_wmma<<<row_tiles, 32, 0, stream>>>(x, W, Wx, N);

    gat_node_prep<<<(N + 255) / 256, 256, 0, stream>>>(
        Wx, att_src, att_dst, s_src, s_dst, e_max, sum_exp, out, N);

    gat_edge_max<<<(E + 255) / 256, 256, 0, stream>>>(
        src, dst, s_src, s_dst, e_max, E);

    gat_edge_sum<<<(E + 255) / 256, 256, 0, stream>>>(
        src, dst, s_src, s_dst, e_max, sum_exp, E);

    unsigned long long tot = (unsigned long long)E * 32ull;
    unsigned int nblk = (unsigned int)((tot + 255ull) / 256ull);
    gat_edge_scatter<<<nblk, 256, 0, stream>>>(
        src, dst, s_src, s_dst, e_max, sum_exp, Wx, out, E);

    gat_elu<<<(N * 32 + 255) / 256, 256, 0, stream>>>(out, N * 32);
}